// KnnDtw_73650099192017
// MI455X (gfx1250) — compile-verified
//
#include <hip/hip_runtime.h>
#include <hip/hip_bf16.h>
#include <stdint.h>

// KNN-DTW for MI455X (gfx1250), compile-only tuning.
// Roofline: 384KB of inputs, ~1G (min,+) ops -> latency/VALU bound. WMMA's
// (x,+) 16x16 tiles cannot express the (min,+) DTW recurrence, so the CDNA5
// wins used here are the data-movement engines instead:
//   * TDM tensor_load_to_lds: one DMA descriptor stages the whole 256x256 f32
//     fit matrix into LDS with hardware padding (pad every 256 dwords by 1
//     dword -> stride-257 rows, bank-conflict-free per-lane reads), tracked
//     by TENSORcnt.
//   * global_load_async_to_lds_b32 (ASYNCcnt) for the per-block query row.
//   * 320KB LDS per WGP: the 260KB resident block is only legal on CDNA5.

#define NQ   128
#define NF   256
#define LEN  256
#define WB   10
#define STR  257            // padded LDS row stride (floats)
#define INFV 3.0e38f

typedef unsigned int v4u __attribute__((ext_vector_type(4)));
typedef int          v8i __attribute__((ext_vector_type(8)));
typedef int          v4i __attribute__((ext_vector_type(4)));

// ---- LDS byte offset of a __shared__ object (generic -> AS(3) -> u32) ------
typedef __attribute__((address_space(3))) const void* lds_cvptr;
__device__ __forceinline__ uint32_t lds_off_u32(const void* p) {
  return (uint32_t)(uintptr_t)(lds_cvptr)p;
}

// ---- CDNA5 async copy: global dword -> LDS (per-lane addresses) ------------
__device__ __forceinline__ void async_g2lds_b32(const float* gp, uint32_t lds_byte) {
  asm volatile("global_load_async_to_lds_b32 %0, %1, off"
               :: "v"(lds_byte), "v"((uint64_t)(uintptr_t)gp)
               : "memory");
}

__device__ __forceinline__ void wait_asynccnt0() {
#if __has_builtin(__builtin_amdgcn_s_wait_asynccnt)
  __builtin_amdgcn_s_wait_asynccnt(0);
#else
  asm volatile("s_wait_asynccnt 0" ::: "memory");
#endif
}

__device__ __forceinline__ void wait_tensorcnt0() {
#if __has_builtin(__builtin_amdgcn_s_wait_tensorcnt)
  __builtin_amdgcn_s_wait_tensorcnt(0);
#else
  asm volatile("s_wait_tensorcnt 0" ::: "memory");
#endif
}

__global__ __launch_bounds__(256)
void knn_dtw_kernel(const float* __restrict__ samples,  // [128][256]
                    const float* __restrict__ fit,      // [256][256]
                    const int*   __restrict__ labels,   // [256]
                    float*       __restrict__ out)      // [128]
{
  __shared__ float sa[LEN];            // query series (broadcast reads)
  __shared__ float sb[NF * STR];       // fit matrix, padded stride 257
  __shared__ float red_v[NF];
  __shared__ int   red_i[NF];

  const int q = blockIdx.x;
  const int f = threadIdx.x;           // one lane per fit series

  // ---------------- stage query row via async-to-LDS (ASYNCcnt) ------------
  async_g2lds_b32(samples + q * LEN + f, lds_off_u32(sa) + (uint32_t)f * 4u);

  // ---------------- stage fit matrix -> LDS (padded stride 257) ------------
#if __has_builtin(__builtin_amdgcn_tensor_load_to_lds)
  if (f < 32) {                         // wave 0 issues one TDM descriptor
    const uint64_t ga = (uint64_t)(uintptr_t)fit;
    const uint32_t lb = lds_off_u32(sb);
    // D# group0: count=1 | lds_addr | global_addr[56:0] | type=2
    v4u g0;
    g0.x = 1u;                                            // count=1, user mode
    g0.y = lb;                                            // lds_addr (bytes)
    g0.z = (uint32_t)ga;                                  // global_addr[31:0]
    g0.w = ((uint32_t)(ga >> 32) & 0x01FFFFFFu) | (2u << 30); // [56:32] | type=2
    // D# group1: data_size=4B, pad_enable, pad_interval=7 (256 dw), pad=1 dw,
    //            tensor 256x256, tile 256x256, dim0_stride=256
    v8i g1;
    g1[0] = (int)((2u << 16) | (1u << 20) | (7u << 22));  // ds=4B|pad_en|intvl
    g1[1] = (int)(256u << 16);   // tensor_dim0[15:0] in bits[63:48]
    g1[2] = (int)(256u << 16);   // tensor_dim1[15:0] in bits[95:80]
    g1[3] = (int)(256u << 16);   // tile_dim0 in bits[127:112]
    g1[4] = (int)256u;           // tile_dim1 in bits[143:128]
    g1[5] = (int)256u;           // tensor_dim0_stride[31:0]
    g1[6] = 0;                   // dim0_stride[47:32]=0, dim1_stride[15:0]=0
    g1[7] = 0;                   // dim1_stride[47:16]=0
    v4i gz4 = {0, 0, 0, 0};      // groups 2/3: tile_dim2=0 -> pure 2D tile
    v8i gz8 = {0, 0, 0, 0, 0, 0, 0, 0};
    // 6-arg form (clang-23 / therock-10.0 headers)
    __builtin_amdgcn_tensor_load_to_lds(g0, g1, gz4, gz4, gz8, 0);
    wait_tensorcnt0();
  }
#else
  {
    const uint32_t bbase = lds_off_u32(sb);
    for (int k = 0; k < NF; ++k) {
      const int n = k * 256 + f;                 // coalesced global reads
      async_g2lds_b32(fit + n, bbase + (uint32_t)(k * STR + f) * 4u);
    }
  }
#endif
  wait_asynccnt0();                    // each wave drains its own ASYNCcnt
  __syncthreads();

  // ---------------- banded DTW DP, band kept in registers ------------------
  // cur[t]  = cost[i][ (i-10) + t ],   t = 0..19   (j in [i-10, i+9])
  // prev[t] = cost[i-1][ (i-11) + t ], t = 0..20
  const float* crow = &sb[f * STR];    // this lane's fit series, stride-257
  float prev[21];
  float cur[20];

  // row 0: cumulative |a0 - b_j| (only j in [0,10] is ever read by row 1)
  {
    const float a0 = sa[0];
    float run = 0.f;
#pragma unroll
    for (int t = 0; t < 21; ++t) {
      const int j = t - WB;            // prev base for i=1 is (0)-10
      if (j >= 0) { run += fabsf(a0 - crow[j]); prev[t] = run; }
      else        { prev[t] = INFV; }
    }
  }
  float colprev = prev[WB];            // cost[0][0]
  const float c0 = crow[0];

  for (int i = 1; i < LEN; ++i) {
    const float ai  = sa[i];
    const float col = colprev + fabsf(ai - c0);      // cost[i][0]
    const int s = (i - WB > 1)   ? (i - WB) : 1;
    const int e = (i + WB < LEN) ? (i + WB) : LEN;   // exclusive
    const int base = i - WB;
    float left = (s == 1) ? col : INFV;              // cost[i][s-1]
#pragma unroll
    for (int t = 0; t < 20; ++t) {
      const int j  = base + t;
      const int jc = j < 0 ? 0 : (j > (LEN - 1) ? (LEN - 1) : j);
      const float bj = crow[jc];
      const float m  = fminf(fminf(prev[t], prev[t + 1]), left);
      const float v  = m + fabsf(ai - bj);
      const bool valid = (j >= s) && (j < e);
      cur[t] = valid ? v : INFV;
      left   = valid ? v : left;
    }
#pragma unroll
    for (int t = 0; t < 20; ++t) prev[t] = cur[t];
    prev[20] = INFV;                   // j = i+10 is outside row i's band
    colprev = col;
  }
  const float dist = prev[WB];         // cost[255][255]

  // ---------------- block argmin (first-index tie-break) -------------------
  red_v[f] = dist;
  red_i[f] = f;
  __syncthreads();
  for (int sft = NF / 2; sft > 0; sft >>= 1) {
    if (f < sft) {
      const float v1 = red_v[f];        const int i1 = red_i[f];
      const float v2 = red_v[f + sft];  const int i2 = red_i[f + sft];
      if ((v2 < v1) || (v2 == v1 && i2 < i1)) { red_v[f] = v2; red_i[f] = i2; }
    }
    __syncthreads();
  }
  if (f == 0) out[q] = (float)labels[red_i[0]];
}

extern "C" void kernel_launch(void* const* d_in, const int* in_sizes, int n_in,
                              void* d_out, int out_size, void* d_ws, size_t ws_size,
                              hipStream_t stream) {
  const float* samples = (const float*)d_in[0];   // [128,256] f32
  const float* fit     = (const float*)d_in[1];   // [256,256] f32
  const int*   labels  = (const int*)d_in[2];     // int labels (0..9)
  float*       out     = (float*)d_out;           // 128 labels as f32
  (void)in_sizes; (void)n_in; (void)out_size; (void)d_ws; (void)ws_size;
  knn_dtw_kernel<<<NQ, NF, 0, stream>>>(samples, fit, labels, out);
}